// CompiledLogicNet_70961449665051
// MI455X (gfx1250) — compile-verified
//
#include <hip/hip_runtime.h>
#include <stdint.h>

// ---------------- problem constants ----------------
#define BATCH   4096
#define NIN     784
#define WIDTH   16000
#define NHID    4
#define NCLS    10
#define NWORDS  (BATCH / 32)          // 128 batch-words (bit-packed batch)
#define NLAYERS (NHID + 1)            // 5 gate layers total
#define GROUP   (WIDTH / NCLS)        // 1600 neurons per class
#define NTILES  (GROUP / 32)          // 50 32-word tiles per class

// ---------------- kernel tuning --------------------
#define T       1024                  // threads per workgroup (32 waves, wave32)
#define CHUNK   2048                  // neurons staged per TDM chunk
#define NCHUNK  ((WIDTH + CHUNK - 1) / CHUNK)   // 8 chunks per layer (last: 1664)
#define GCHUNKS (NLAYERS * NCHUNK)              // 40 chunks total

// ---------------- LDS layout (32-bit words) --------
#define OFF_HA      0                 // 16000 words : layer ping
#define OFF_HB      16000             // 16000 words : layer pong
#define OFF_XCOL    32000             // 784 words   : packed input column
#define OFF_STAGE   32784             // double-buffered idx/neg staging (reused as partials)
#define STAGE_WORDS 5120              // ia[2048] + ib[2048] + neg(4096B = 1024w)
#define SMEM_WORDS  (OFF_STAGE + 2 * STAGE_WORDS)   // 43024 words
#define SMEM_BYTES  (SMEM_WORDS * 4)                // 172096 B (< 320 KB WGP LDS)

typedef int   v4i __attribute__((ext_vector_type(4)));
typedef int   v8i __attribute__((ext_vector_type(8)));
typedef float v2f __attribute__((ext_vector_type(2)));
typedef float v8f __attribute__((ext_vector_type(8)));

// ---- CDNA5 Tensor Data Mover: 1D tile, global -> LDS (TENSORcnt) ----
// D# per CDNA5 ISA (2 SGPR groups, VADDR2/3 = NULL)
__device__ __forceinline__ void tdm_load_1d(uint32_t lds_byte_off, const void* gp,
                                            uint32_t elems, uint32_t ds_log2) {
  unsigned long long ga = (unsigned long long)gp;
  v4i g0;
  g0[0] = 1;                                             // count=1, user mode
  g0[1] = (int)lds_byte_off;                             // lds_addr
  g0[2] = (int)(uint32_t)ga;                             // global_addr[31:0]
  g0[3] = (int)(((uint32_t)(ga >> 32) & 0x01FFFFFFu)     // global_addr[56:32]
                | 0x80000000u);                          // type=2 ("image")
  v8i g1;
  g1[0] = (int)(ds_log2 << 16);                          // data_size, mask=0, no opts
  g1[1] = (int)(elems << 16);                            // tensor_dim0[15:0]
  g1[2] = (int)(1u << 16);                               // tensor_dim0 hi=0 | tensor_dim1=1
  g1[3] = (int)(elems << 16);                            // tile_dim0 = elems
  g1[4] = 0;                                             // tile_dim1/2 unused
  g1[5] = (int)elems;                                    // tensor_dim0_stride
  g1[6] = 0;
  g1[7] = 0;
  asm volatile("tensor_load_to_lds %0, %1" :: "s"(g0), "s"(g1) : "memory");
}

// ---- wave32 32x32 bit transpose step: xor-lane exchange + block swap ----
#define TR_STEP(J, LO)                                                          \
  {                                                                             \
    uint32_t y = (uint32_t)__builtin_amdgcn_ds_swizzle((int)xw,                 \
                                                       (((J) << 10) | 0x1f));   \
    xw = (ln & (J)) ? ((xw & ~(LO)) | ((y & ~(LO)) >> (J)))                     \
                    : ((xw & (LO)) | ((y & (LO)) << (J)));                      \
  }

// ---------------- kernel 1: bit-pack x over the batch dim ----------------
__global__ void pack_x_kernel(const uint8_t* __restrict__ x,
                              uint32_t* __restrict__ px) {
  int i = blockIdx.x * blockDim.x + threadIdx.x;
  if (i >= NWORDS * (NIN / 4)) return;
  const int Wd = i / (NIN / 4), c4 = i % (NIN / 4);
  const uint8_t* xp = x + (size_t)Wd * 32 * NIN + (size_t)c4 * 4;
  uint32_t w0 = 0, w1 = 0, w2 = 0, w3 = 0;
#pragma unroll
  for (int p = 0; p < 32; ++p) {
    const uint32_t d = *(const uint32_t*)(xp + (size_t)p * NIN);
    w0 |= (d & 1u) << p;
    w1 |= ((d >> 8) & 1u) << p;
    w2 |= ((d >> 16) & 1u) << p;
    w3 |= ((d >> 24) & 1u) << p;
  }
  *(uint4*)(px + (size_t)Wd * NIN + (size_t)c4 * 4) = make_uint4(w0, w1, w2, w3);
}

// ---------------- kernel 2: LDS-resident logic net, one WG per batch-word ----
__global__ __launch_bounds__(T) void logic_net_kernel(
    const int* __restrict__ idx_a0, const int* __restrict__ idx_b0,
    const uint8_t* __restrict__ neg0,
    const int* __restrict__ idx_a, const int* __restrict__ idx_b,
    const uint8_t* __restrict__ neg,
    const uint32_t* __restrict__ packed_x,
    float* __restrict__ out) {
  extern __shared__ uint32_t smem[];
  const int t = threadIdx.x;
  const int W = blockIdx.x;

  uint32_t* hA   = smem + OFF_HA;
  uint32_t* hB   = smem + OFF_HB;
  uint32_t* xcol = smem + OFF_XCOL;
  const uint32_t lds0 = (uint32_t)(uintptr_t)smem;   // LDS byte offset of dyn base

  // wave 0 drives all TDM staging: 3 descriptors per chunk (ia / ib / neg)
  auto issue_stage = [&](int g) {
    const int l = g / NCHUNK, c = g % NCHUNK;
    const int base = c * CHUNK;
    int cnt = WIDTH - base; if (cnt > CHUNK) cnt = CHUNK;
    const int*     pa = (l == 0) ? idx_a0 : idx_a + (size_t)(l - 1) * WIDTH;
    const int*     pb = (l == 0) ? idx_b0 : idx_b + (size_t)(l - 1) * WIDTH;
    const uint8_t* pn = (l == 0) ? neg0   : neg   + (size_t)(l - 1) * WIDTH * 2;
    const uint32_t sb = lds0 + (OFF_STAGE + (g & 1) * STAGE_WORDS) * 4;
    tdm_load_1d(sb,         pa + base,     (uint32_t)cnt,     2);  // dwords
    tdm_load_1d(sb + 8192,  pb + base,     (uint32_t)cnt,     2);  // dwords
    tdm_load_1d(sb + 16384, pn + 2 * base, (uint32_t)cnt * 2, 0);  // bytes
  };

  // ---- prologue: TDM xcol (1) + chunk 0 staging (3), wave 0 only
  if (t < 32) {
    tdm_load_1d(lds0 + OFF_XCOL * 4, packed_x + (size_t)W * NIN, NIN, 2);
    issue_stage(0);
  }

  // ---- 5 layers x 8 chunks; TENSORcnt-pipelined staging, distance 1 ----
  for (int g = 0; g < GCHUNKS; ++g) {
    if (t < 32) {
      if (g + 1 < GCHUNKS) {
        issue_stage(g + 1);                       // 3 newest in flight
        __builtin_amdgcn_s_wait_tensorcnt(3);     // all older (chunk g, xcol) done
      } else {
        __builtin_amdgcn_s_wait_tensorcnt(0);
      }
    }
    __syncthreads();                              // chunk g staged for all waves

    const int l = g / NCHUNK, c = g % NCHUNK;
    const uint32_t* src = (l == 0) ? xcol : ((l & 1) ? hA : hB);
    uint32_t*       dst = (l & 1) ? hB : hA;      // L0->hA, L1->hB, ... L4->hA

    const uint32_t* sia = smem + OFF_STAGE + (g & 1) * STAGE_WORDS;
    const uint32_t* sib = sia + CHUNK;
    const uint16_t* snf = (const uint16_t*)(sia + 2 * CHUNK);

    const int base = c * CHUNK;
    int cnt = WIDTH - base; if (cnt > CHUNK) cnt = CHUNK;
    for (int k = t; k < cnt; k += T) {
      const uint32_t a  = src[sia[k]];            // ds gather
      const uint32_t b  = src[sib[k]];            // ds gather
      const uint32_t nf = snf[k];                 // {nb,na} bytes
      const uint32_t ma = 0u - (nf & 1u);
      const uint32_t mb = 0u - ((nf >> 8) & 1u);
      dst[base + k] = (a ^ ma) & (b ^ mb);        // the logic gate, 32 batch/op
    }
    __syncthreads();                              // staging buf free; layer done
  }

  // ---- GroupSum stage 1: 32x32 bit transpose (ds_swizzle) + popcount ----
  // 30 waves, wave = (class, third); lane ln ends up owning batch bit ln.
  float* part = (float*)(smem + OFF_STAGE);       // [NCLS][3][32] partials (staging dead)
  const int wv = t >> 5, ln = t & 31;
  if (wv < NCLS * 3) {
    const int c = wv / 3, s = wv % 3;
    const int tb = s * 17;
    const int te = (s == 2) ? NTILES : tb + 17;   // 17/17/16 tiles of 32 words
    uint32_t acc = 0;
    for (int tl = tb; tl < te; ++tl) {
      uint32_t xw = hA[c * GROUP + tl * 32 + ln]; // lane ln: word of neuron tl*32+ln
      TR_STEP(16, 0x0000FFFFu)
      TR_STEP(8,  0x00FF00FFu)
      TR_STEP(4,  0x0F0F0F0Fu)
      TR_STEP(2,  0x33333333u)
      TR_STEP(1,  0x55555555u)
      acc += (uint32_t)__popc(xw);                // 32 neurons of batch bit ln at once
    }
    part[(c * 3 + s) * 32 + ln] = (float)acc;     // exact: acc <= 1600
  }
  __syncthreads();

  // ---- GroupSum stage 2: segment combine as 16x4 * 4x16 WMMA matvec ----
  // logits[o] = sum_s part[s][o], o = c*32+p in [0,320). Wave w handles 16 outputs.
  // A layout (f32 16x4): lane L<16 -> row L, K=0/1; lane L+16 -> row L, K=2/3.
  // B = ones with B[3][*] = 0  =>  every D column holds the row sum.
  if (wv < 20) {
    const int m  = ln & 15;                       // output row within this wmma
    const int o  = wv * 16 + m;                   // global output id
    const int c  = o >> 5, p = o & 31;
    const bool hi = (ln >= 16);
    v2f a;
    a.x = hi ? part[(c * 3 + 2) * 32 + p] : part[(c * 3 + 0) * 32 + p]; // K=2 : K=0
    a.y = hi ? 0.0f                       : part[(c * 3 + 1) * 32 + p]; // K=3 : K=1
    v2f b;
    b.x = 1.0f;                                   // B rows K=0, K=2
    b.y = hi ? 0.0f : 1.0f;                       // B rows K=3, K=1
    v8f cacc = {};
    v8f d = __builtin_amdgcn_wmma_f32_16x16x4_f32(false, a, false, b,
                                                  (short)0, cacc, false, false);
    if (ln == 0 || ln == 16) {                    // lane0: M=0..7, lane16: M=8..15
#pragma unroll
      for (int i = 0; i < 8; ++i) {
        const int o2 = wv * 16 + (hi ? 8 : 0) + i;
        const int c2 = o2 >> 5, p2 = o2 & 31;
        out[(size_t)(W * 32 + p2) * NCLS + c2] = d[i];
      }
    }
  }
}

// ---------------- host launcher ----------------
extern "C" void kernel_launch(void* const* d_in, const int* in_sizes, int n_in,
                              void* d_out, int out_size, void* d_ws, size_t ws_size,
                              hipStream_t stream) {
  // setup_inputs() order: x, idx_a0, idx_b0, neg0, idx_a, idx_b, neg
  // bool arrays assumed JAX-native 1-byte storage.
  const uint8_t* x      = (const uint8_t*)d_in[0];
  const int*     idx_a0 = (const int*)d_in[1];
  const int*     idx_b0 = (const int*)d_in[2];
  const uint8_t* neg0   = (const uint8_t*)d_in[3];
  const int*     idx_a  = (const int*)d_in[4];
  const int*     idx_b  = (const int*)d_in[5];
  const uint8_t* neg    = (const uint8_t*)d_in[6];
  uint32_t* px  = (uint32_t*)d_ws;          // [NWORDS][NIN] packed input
  float*    out = (float*)d_out;            // [BATCH][NCLS] logits (int-valued)

  (void)in_sizes; (void)n_in; (void)out_size; (void)ws_size;

  hipFuncSetAttribute((const void*)logic_net_kernel,
                      hipFuncAttributeMaxDynamicSharedMemorySize, SMEM_BYTES);

  const int total = NWORDS * (NIN / 4);
  pack_x_kernel<<<(total + 255) / 256, 256, 0, stream>>>(x, px);
  logic_net_kernel<<<NWORDS, T, SMEM_BYTES, stream>>>(
      idx_a0, idx_b0, neg0, idx_a, idx_b, neg, px, out);
}